// HashEncoder_36163624633055
// MI455X (gfx1250) — compile-verified
//
#include <hip/hip_runtime.h>
#include <cmath>

// Instant-NGP hash-grid encode, MI455X (gfx1250, wave32).
// 16 levels x 8-corner gathers from 2^19-entry tables (4 MB/level, 64 MB total:
// fully L2-resident on the 192 MB L2). Gather-throughput bound; streaming
// traffic ~344 MB => ~15us floor at 23.3 TB/s.
//
// This revision fully unrolls the level loop and accumulates the 32 output
// floats in registers, then writes the 128 B row with 8 back-to-back b128 NT
// stores so each wave completes 32 full cachelines in 8 instructions (ideal
// write-combining; the earlier per-level 8 B stores left lines partially
// written across the whole loop). Full unroll also lets the scheduler keep
// multiple levels' gather batches in flight against L2 latency.

#define NLVL   16
#define LOG2T  19
#define TSIZE  (1u << LOG2T)
#define TMASK  (TSIZE - 1u)

typedef float v2f __attribute__((ext_vector_type(2)));
typedef float v4f __attribute__((ext_vector_type(4)));

struct ResPack { int r[NLVL]; };

__global__ __launch_bounds__(256) void hashgrid_fwd(
    const float* __restrict__ pos,     // [N,3] f32
    const float* __restrict__ tables,  // [16, 2^19, 2] f32
    float* __restrict__ out,           // [N, 32] f32
    int n, ResPack rp)
{
    const int i = blockIdx.x * 256 + threadIdx.x;
    if (i >= n) return;

    // Warm L2 for the position stream ~4 blocks ahead (gfx1250 global_prefetch_b8).
    if (i + 4 * 256 < n)
        __builtin_prefetch(pos + (size_t)3 * (i + 4 * 256), 0, 1);

    // Positions are a read-once stream: non-temporal (compiler fuses to b96 NT).
    const float px = __builtin_nontemporal_load(pos + 3 * (size_t)i + 0);
    const float py = __builtin_nontemporal_load(pos + 3 * (size_t)i + 1);
    const float pz = __builtin_nontemporal_load(pos + 3 * (size_t)i + 2);

    const unsigned P1 = 2654435761u, P2 = 805459861u, P3 = 3674653429u;

    float acc0[NLVL];  // feature 0 per level
    float acc1[NLVL];  // feature 1 per level

#pragma unroll
    for (int l = 0; l < NLVL; ++l) {
        const int   res = rp.r[l];
        const int   rm  = res - 1;
        const float s   = (float)rm;

        const float sx = px * s, sy = py * s, sz = pz * s;
        const float fx = floorf(sx), fy = floorf(sy), fz = floorf(sz);
        const float wx = sx - fx,  wy = sy - fy,  wz = sz - fz;

        int x0 = min(max((int)fx, 0), rm);
        int y0 = min(max((int)fy, 0), rm);
        int z0 = min(max((int)fz, 0), rm);
        int x1 = min(x0 + 1, rm);
        int y1 = min(y0 + 1, rm);
        int z1 = min(z0 + 1, rm);

        // Spatial hash with uint32 wraparound; modulus is 2^19 -> mask.
        const unsigned hx0 = (unsigned)x0 * P1, hx1 = (unsigned)x1 * P1;
        const unsigned hy0 = (unsigned)y0 * P2, hy1 = (unsigned)y1 * P2;
        const unsigned hz0 = (unsigned)z0 * P3, hz1 = (unsigned)z1 * P3;

        const unsigned i000 = (hx0 ^ hy0 ^ hz0) & TMASK;
        const unsigned i001 = (hx0 ^ hy0 ^ hz1) & TMASK;
        const unsigned i010 = (hx0 ^ hy1 ^ hz0) & TMASK;
        const unsigned i011 = (hx0 ^ hy1 ^ hz1) & TMASK;
        const unsigned i100 = (hx1 ^ hy0 ^ hz0) & TMASK;
        const unsigned i101 = (hx1 ^ hy0 ^ hz1) & TMASK;
        const unsigned i110 = (hx1 ^ hy1 ^ hz0) & TMASK;
        const unsigned i111 = (hx1 ^ hy1 ^ hz1) & TMASK;

        // 8 independent b64 gathers, all in flight before any use (L2-resident).
        const v2f* __restrict__ tbl = (const v2f*)tables + (size_t)l * TSIZE;
        const v2f c000 = tbl[i000];
        const v2f c001 = tbl[i001];
        const v2f c010 = tbl[i010];
        const v2f c011 = tbl[i011];
        const v2f c100 = tbl[i100];
        const v2f c101 = tbl[i101];
        const v2f c110 = tbl[i110];
        const v2f c111 = tbl[i111];

        // Trilinear corner weights (factored: 3 + 4 + 8 muls).
        const float wx0 = 1.f - wx, wy0 = 1.f - wy, wz0 = 1.f - wz;
        const float a00 = wy0 * wz0, a01 = wy0 * wz, a10 = wy * wz0, a11 = wy * wz;
        const float w000 = wx0 * a00, w001 = wx0 * a01, w010 = wx0 * a10, w011 = wx0 * a11;
        const float w100 = wx  * a00, w101 = wx  * a01, w110 = wx  * a10, w111 = wx  * a11;

        float f0 = c000.x * w000;
        f0 = fmaf(c001.x, w001, f0);
        f0 = fmaf(c010.x, w010, f0);
        f0 = fmaf(c011.x, w011, f0);
        f0 = fmaf(c100.x, w100, f0);
        f0 = fmaf(c101.x, w101, f0);
        f0 = fmaf(c110.x, w110, f0);
        f0 = fmaf(c111.x, w111, f0);

        float f1 = c000.y * w000;
        f1 = fmaf(c001.y, w001, f1);
        f1 = fmaf(c010.y, w010, f1);
        f1 = fmaf(c011.y, w011, f1);
        f1 = fmaf(c100.y, w100, f1);
        f1 = fmaf(c101.y, w101, f1);
        f1 = fmaf(c110.y, w110, f1);
        f1 = fmaf(c111.y, w111, f1);

        acc0[l] = f0;
        acc1[l] = f1;
    }

    // Write the full 128 B output row with 8 consecutive b128 NT stores:
    // each wave fills 32 complete cachelines back-to-back (ideal WC), and the
    // NT hint keeps the 256 MB output stream from evicting the L2-resident
    // hash tables.
    float* orow = out + (size_t)i * (NLVL * 2);
#pragma unroll
    for (int j = 0; j < NLVL / 2; ++j) {
        v4f r;
        r.x = acc0[2 * j];
        r.y = acc1[2 * j];
        r.z = acc0[2 * j + 1];
        r.w = acc1[2 * j + 1];
        __builtin_nontemporal_store(r, (v4f*)(orow + 4 * j));
    }
}

extern "C" void kernel_launch(void* const* d_in, const int* in_sizes, int n_in,
                              void* d_out, int out_size, void* d_ws, size_t ws_size,
                              hipStream_t stream) {
    const float* pos    = (const float*)d_in[0];   // [N,3]
    const float* tables = (const float*)d_in[1];   // [16, 2^19, 2]
    float*       out    = (float*)d_out;           // [N, 32]
    const int n = in_sizes[0] / 3;

    // Per-level resolutions: replicate the reference's double-precision math
    // (b = exp((ln 2048 - ln 16)/15); res_l = min(int(16 * b^l), 2048)).
    ResPack rp;
    const double b = std::exp((std::log(2048.0) - std::log(16.0)) / 15.0);
    for (int l = 0; l < NLVL; ++l) {
        const double v = 16.0 * std::pow(b, (double)l);
        int r = (int)v;
        if (r > 2048) r = 2048;
        rp.r[l] = r;
    }

    const int blocks = (n + 255) / 256;
    hipLaunchKernelGGL(hashgrid_fwd, dim3(blocks), dim3(256), 0, stream,
                       pos, tables, out, n, rp);
}